// CayleyConvED2_67242007986418
// MI455X (gfx1250) — compile-verified
//
#include <hip/hip_runtime.h>
#include <hip/hip_bf16.h>
#include <math.h>

// ---------------------------------------------------------------------------
// CayleyConvED2 for MI455X (gfx1250, wave32, WMMA + TDM).
//
// Pipeline:
//  1. k_xfft   : rfft2(x) -> xfft (F=544, C=256, B=32) split re/im
//  2. k_norm   : ||wfft||_F^2 (9-term twiddle sums, block-reduce + atomic)
//  3. k_scale  : scale = alpha / ||wfft||
//  4. per frequency chunk:
//     k_buildA : A = scale * (wfft - wfft^H)           (F,256,256) complex
//     k_capply x4 (Horner/Neumann):  wx = x - 2A(x - A(x - A(x - A x)))
//                batched complex GEMM 256x32x256 via v_wmma_f32_16x16x4_f32,
//                A/B panels staged into LDS by the Tensor Data Mover
//                (tensor_load_to_lds + s_wait_tensorcnt), with D# padding
//                fields reproducing the bank-conflict-free LDS strides.
//  5. k_mixH   : frequency-domain 1x1 conv  wx2 = H * wx   (real WMMA GEMM)
//  6. k_irfft  : irfft2 + bias -> out (B,256,32,32)
// ---------------------------------------------------------------------------

typedef float v2f __attribute__((ext_vector_type(2)));
typedef float v8f __attribute__((ext_vector_type(8)));
typedef unsigned int u32x4 __attribute__((ext_vector_type(4)));
typedef unsigned int u32x8 __attribute__((ext_vector_type(8)));

#define C_DIM 256
#define B_DIM 32
#define NFREQ 544           // 32 * 17
#define KC    64            // K-chunk staged in LDS
#define TWOPI_32 0.19634954084936207f   // 2*pi/32

// ---------------------------------------------------------------------------
// TDM: issue a 2D tile load global->LDS via a Tensor DMA Descriptor (D#).
//   - group0: count=1 | lds_addr | global_addr[56:0] | type=2
//   - group1: data_size=4B, pad_enable, pad_interval/amount, tile dims,
//             dim0 row stride (elements).  2-group form => up to 2D tensors.
// Issued by one wave only (TDM ignores EXEC); completion via TENSORcnt.
// ---------------------------------------------------------------------------
__device__ __forceinline__ void tdm_load_2d(const float* gptr, unsigned lds_byte,
                                            unsigned tile_d0, unsigned tile_d1,
                                            unsigned row_stride,
                                            unsigned pad_interval,
                                            unsigned pad_amount) {
  unsigned long long ga = (unsigned long long)(uintptr_t)gptr;
  u32x4 g0;
  g0[0] = 1u;                                              // count=1 (user D#)
  g0[1] = lds_byte;                                        // lds_addr
  g0[2] = (unsigned)ga;                                    // global_addr[31:0]
  g0[3] = (unsigned)((ga >> 32) & 0x01ffffffu) | (2u << 30);  // [56:32] | type=2
  u32x8 g1;
  g1[0] = (2u << 16) | (1u << 20) | (pad_interval << 22) | (pad_amount << 25);
  g1[1] = 0u;                        // tensor_dim0 lo16 (dim0 = 0x40000000)
  g1[2] = 0x4000u;                   // tensor_dim0 hi16 | tensor_dim1 lo16
  g1[3] = 0x4000u | (tile_d0 << 16); // tensor_dim1 hi16 | tile_dim0
  g1[4] = tile_d1;                   // tile_dim1 | tile_dim2=0
  g1[5] = row_stride;                // tensor_dim0_stride lo32 (elements)
  g1[6] = 0u;
  g1[7] = 0u;
  asm volatile("tensor_load_to_lds %0, %1" :: "s"(g0), "s"(g1) : "memory");
}

// ---------------------------------------------------------------------------
// 9-term twiddle sum: wfft[u,v,i,j] = sum_{d,e in {-1,0,1}} w[p=d+1][q=e+1]
//                                      * exp(2*pi*i*(u*d + v*e)/32)
// (this already folds in the conj() and the fft_shift(-s) phase, s=1)
// ---------------------------------------------------------------------------
__device__ __forceinline__ void wfft9(const float* __restrict__ wp,
                                      float cu, float su, float cv, float sv,
                                      float& outr, float& outi) {
  float eur[3] = {cu, 1.0f, cu};
  float eui[3] = {-su, 0.0f, su};
  float evr[3] = {cv, 1.0f, cv};
  float evi[3] = {-sv, 0.0f, sv};
  float sr = 0.0f, si = 0.0f;
#pragma unroll
  for (int p = 0; p < 3; ++p) {
#pragma unroll
    for (int q = 0; q < 3; ++q) {
      float tr = eur[p] * evr[q] - eui[p] * evi[q];
      float ti = eur[p] * evi[q] + eui[p] * evr[q];
      float wv = wp[p * 3 + q];
      sr += wv * tr;
      si += wv * ti;
    }
  }
  outr = sr;
  outi = si;
}

// ---------------------------------------------------------------------------
// rfft2 of x, one (b,c) image per workgroup.  Output layout (F, C, B).
// ---------------------------------------------------------------------------
__global__ __launch_bounds__(256) void k_xfft(const float* __restrict__ x,
                                              float* __restrict__ xr,
                                              float* __restrict__ xi) {
  const int b = blockIdx.x >> 8;
  const int c = blockIdx.x & 255;
  __shared__ float xs[32][33];
  __shared__ float twc[32], tws[32];
  __shared__ float gr[32][17], gi[32][17];
  const int tid = threadIdx.x;
  if (tid < 32) {
    float s, co;
    __sincosf(TWOPI_32 * (float)tid, &s, &co);
    twc[tid] = co;
    tws[tid] = s;
  }
  const float* xp = x + (size_t)(b * 256 + c) * 1024;
  for (int it = tid; it < 1024; it += 256) xs[it >> 5][it & 31] = xp[it];
  __syncthreads();
  // stage 1: along q (last axis): g[p][v] = sum_q x[p][q] e^{-2pi i v q/32}
  for (int idx = tid; idx < 544; idx += 256) {
    int p = idx / 17, v = idx % 17;
    float ar = 0.0f, ai = 0.0f;
#pragma unroll 4
    for (int q = 0; q < 32; ++q) {
      int t = (v * q) & 31;
      float xv = xs[p][q];
      ar += xv * twc[t];
      ai -= xv * tws[t];
    }
    gr[p][v] = ar;
    gi[p][v] = ai;
  }
  __syncthreads();
  // stage 2: along p: X[u][v] = sum_p g[p][v] e^{-2pi i u p/32}
  for (int idx = tid; idx < 544; idx += 256) {
    int u = idx / 17, v = idx % 17;
    float ar = 0.0f, ai = 0.0f;
#pragma unroll 4
    for (int p = 0; p < 32; ++p) {
      int t = (u * p) & 31;
      float cc = twc[t], ss = tws[t];
      float grv = gr[p][v], giv = gi[p][v];
      ar += grv * cc + giv * ss;
      ai += giv * cc - grv * ss;
    }
    size_t o = (size_t)idx * (C_DIM * B_DIM) + (size_t)c * B_DIM + b;
    xr[o] = ar;
    xi[o] = ai;
  }
}

// ---------------------------------------------------------------------------
// ||wfft||_F^2 accumulation.
// ---------------------------------------------------------------------------
__global__ __launch_bounds__(256) void k_norm(const float* __restrict__ w,
                                              float* __restrict__ norm2) {
  __shared__ float red[256];
  size_t gid = (size_t)blockIdx.x * 256 + threadIdx.x;  // over F*C*C
  int f = (int)(gid >> 16);
  int ij = (int)(gid & 65535);
  int i = ij >> 8, j = ij & 255;
  int u = f / 17, v = f % 17;
  float su, cu, sv, cv;
  __sincosf(TWOPI_32 * (float)u, &su, &cu);
  __sincosf(TWOPI_32 * (float)v, &sv, &cv);
  float wr, wi;
  wfft9(w + ((size_t)(i * 256 + j)) * 9, cu, su, cv, sv, wr, wi);
  red[threadIdx.x] = wr * wr + wi * wi;
  __syncthreads();
  for (int s = 128; s > 0; s >>= 1) {
    if (threadIdx.x < s) red[threadIdx.x] += red[threadIdx.x + s];
    __syncthreads();
  }
  if (threadIdx.x == 0) atomicAdd(norm2, red[0]);
}

__global__ void k_scale(const float* __restrict__ alpha,
                        const float* __restrict__ norm2,
                        float* __restrict__ scale) {
  scale[0] = alpha[0] * rsqrtf(fmaxf(norm2[0], 1e-30f));
}

// ---------------------------------------------------------------------------
// A = scale * (wfft - wfft^H) for a chunk of frequencies [f0, f0+nf)
// ---------------------------------------------------------------------------
__global__ __launch_bounds__(256) void k_buildA(const float* __restrict__ w,
                                                const float* __restrict__ scale_p,
                                                float* __restrict__ A_re,
                                                float* __restrict__ A_im,
                                                int f0, int nf) {
  size_t gid = (size_t)blockIdx.x * 256 + threadIdx.x;
  if (gid >= (size_t)nf * 65536) return;
  int fl = (int)(gid >> 16);
  int ij = (int)(gid & 65535);
  int i = ij >> 8, j = ij & 255;
  int fa = f0 + fl;
  int u = fa / 17, v = fa % 17;
  float su, cu, sv, cv;
  __sincosf(TWOPI_32 * (float)u, &su, &cu);
  __sincosf(TWOPI_32 * (float)v, &sv, &cv);
  float rij, iij, rji, iji;
  wfft9(w + ((size_t)(i * 256 + j)) * 9, cu, su, cv, sv, rij, iij);
  wfft9(w + ((size_t)(j * 256 + i)) * 9, cu, su, cv, sv, rji, iji);
  float s = scale_p[0];
  A_re[gid] = s * (rij - rji);   // Re(Wij - conj(Wji))
  A_im[gid] = s * (iij + iji);   // Im(Wij - conj(Wji))
}

// ---------------------------------------------------------------------------
// Batched complex GEMM+AXPY:  out[f] = xfft[f_abs] + coef * (A[f] * in[f])
// A: (256,256) complex per freq, in/out: (256,32) complex per freq.
// One workgroup = one freq x one 64-row M-block; 8 waves each own a 16x16
// complex tile (4 mtiles x 2 ntiles).  K staged through LDS in chunks of 64
// by the Tensor Data Mover (wave 0 issues, TENSORcnt + barrier publish).
// WMMA: v_wmma_f32_16x16x4_f32, 4 per complex K-step.
// ---------------------------------------------------------------------------
__global__ __launch_bounds__(256) void k_capply(
    const float* __restrict__ A_re, const float* __restrict__ A_im,
    const float* __restrict__ xb_re, const float* __restrict__ xb_im,
    const float* __restrict__ in_re, const float* __restrict__ in_im,
    float* __restrict__ out_re, float* __restrict__ out_im,
    int f0, int in_abs, int out_abs, float coef) {
  __shared__ float As_re[64][68];   // 64 data DW + 4 pad DW per row
  __shared__ float As_im[64][68];
  __shared__ float Bs_re[KC][33];   // 32 data DW + 1 pad DW per row
  __shared__ float Bs_im[KC][33];

  const int fl = blockIdx.x;
  const int fa = f0 + fl;
  const int i0 = blockIdx.y * 64;
  const int tid = (int)threadIdx.x;
  const int lane = tid & 31;
  const int wave = tid >> 5;
  const int mtile = wave & 3;
  const int ntile = wave >> 2;
  const int m = lane & 15;
  const int hi = lane >> 4;

  const size_t a_base = (size_t)fl * (C_DIM * C_DIM) + (size_t)i0 * C_DIM;
  const size_t in_base = (size_t)(in_abs ? fa : fl) * (C_DIM * B_DIM);
  const size_t out_base = (size_t)(out_abs ? fa : fl) * (C_DIM * B_DIM);
  const size_t x_base = (size_t)fa * (C_DIM * B_DIM);

  v8f cre = {};
  v8f cim = {};

  for (int kc = 0; kc < C_DIM; kc += KC) {
    __syncthreads();   // previous iteration done reading LDS
    if (wave == 0) {
      // TDM: A panel 64x64 (row stride 256), pad 4 DW / 64 DW -> [64][68]
      tdm_load_2d(A_re + a_base + kc, (unsigned)(uintptr_t)&As_re[0][0],
                  KC, 64, C_DIM, 5u, 3u);
      tdm_load_2d(A_im + a_base + kc, (unsigned)(uintptr_t)&As_im[0][0],
                  KC, 64, C_DIM, 5u, 3u);
      // TDM: B panel 64x32 (row stride 32), pad 1 DW / 32 DW -> [64][33]
      tdm_load_2d(in_re + in_base + (size_t)kc * B_DIM,
                  (unsigned)(uintptr_t)&Bs_re[0][0], B_DIM, KC, B_DIM, 4u, 0u);
      tdm_load_2d(in_im + in_base + (size_t)kc * B_DIM,
                  (unsigned)(uintptr_t)&Bs_im[0][0], B_DIM, KC, B_DIM, 4u, 0u);
      __builtin_amdgcn_s_wait_tensorcnt(0);
    }
    if (kc + KC < C_DIM) {  // global_prefetch_b8 of next A panel
      __builtin_prefetch(A_re + a_base + (size_t)(tid >> 2) * C_DIM + kc + KC, 0, 1);
    }
    __syncthreads();   // publish TDM-written LDS to all waves

    const int mr = mtile * 16 + m;
    const int nn = ntile * 16 + m;
#pragma unroll 4
    for (int ks = 0; ks < KC / 4; ++ks) {
      int kk = ks * 4 + 2 * hi;  // A frag: K = 2*hi + vgpr  (ISA 16x4 layout)
      v2f are = *(const v2f*)&As_re[mr][kk];
      v2f aim = *(const v2f*)&As_im[mr][kk];
      v2f aimn = -aim;
      v2f bre, bim;             // B frag: row K = 2*hi + vgpr, col N = lane&15
      bre.x = Bs_re[kk + 0][nn];
      bre.y = Bs_re[kk + 1][nn];
      bim.x = Bs_im[kk + 0][nn];
      bim.y = Bs_im[kk + 1][nn];
      cre = __builtin_amdgcn_wmma_f32_16x16x4_f32(false, are,  false, bre, (short)0, cre, false, false);
      cre = __builtin_amdgcn_wmma_f32_16x16x4_f32(false, aimn, false, bim, (short)0, cre, false, false);
      cim = __builtin_amdgcn_wmma_f32_16x16x4_f32(false, are,  false, bim, (short)0, cim, false, false);
      cim = __builtin_amdgcn_wmma_f32_16x16x4_f32(false, aim,  false, bre, (short)0, cim, false, false);
    }
  }

  // D layout: element (M = r + 8*hi, N = lane&15) in vgpr r
  const int col = ntile * 16 + (lane & 15);
#pragma unroll
  for (int r = 0; r < 8; ++r) {
    int row = i0 + mtile * 16 + r + 8 * hi;
    size_t xo = x_base + (size_t)row * B_DIM + col;
    size_t oo = out_base + (size_t)row * B_DIM + col;
    out_re[oo] = xb_re[xo] + coef * cre[r];
    out_im[oo] = xb_im[xo] + coef * cim[r];
  }
}

// ---------------------------------------------------------------------------
// Frequency-domain 1x1 conv: out[f] = H (256x256, real) * wx[f] (256x32 cplx)
// (classic vectorized global->LDS staging path)
// ---------------------------------------------------------------------------
__global__ __launch_bounds__(256) void k_mixH(const float* __restrict__ Hm,
                                              const float* __restrict__ in_re,
                                              const float* __restrict__ in_im,
                                              float* __restrict__ out_re,
                                              float* __restrict__ out_im) {
  __shared__ float Hs[64][68];
  __shared__ float Bs_re[KC][33];
  __shared__ float Bs_im[KC][33];

  const int f = blockIdx.x;
  const int o0 = blockIdx.y * 64;
  const int tid = (int)threadIdx.x;
  const int lane = tid & 31;
  const int wave = tid >> 5;
  const int mtile = wave & 3;
  const int ntile = wave >> 2;
  const int m = lane & 15;
  const int hi = lane >> 4;

  const size_t h_base = (size_t)o0 * C_DIM;
  const size_t f_base = (size_t)f * (C_DIM * B_DIM);

  v8f cre = {};
  v8f cim = {};

  for (int kc = 0; kc < C_DIM; kc += KC) {
    __syncthreads();
    for (int it = tid; it < 64 * (KC / 4); it += 256) {
      int r = it / (KC / 4);
      int c4 = (it % (KC / 4)) << 2;
      float4 vh = *(const float4*)(Hm + h_base + (size_t)r * C_DIM + kc + c4);
      Hs[r][c4 + 0] = vh.x; Hs[r][c4 + 1] = vh.y;
      Hs[r][c4 + 2] = vh.z; Hs[r][c4 + 3] = vh.w;
    }
    for (int it = tid; it < KC * 8; it += 256) {
      int r = it >> 3;
      int c4 = (it & 7) << 2;
      float4 vr = *(const float4*)(in_re + f_base + (size_t)(kc + r) * B_DIM + c4);
      float4 vi = *(const float4*)(in_im + f_base + (size_t)(kc + r) * B_DIM + c4);
      Bs_re[r][c4 + 0] = vr.x; Bs_re[r][c4 + 1] = vr.y;
      Bs_re[r][c4 + 2] = vr.z; Bs_re[r][c4 + 3] = vr.w;
      Bs_im[r][c4 + 0] = vi.x; Bs_im[r][c4 + 1] = vi.y;
      Bs_im[r][c4 + 2] = vi.z; Bs_im[r][c4 + 3] = vi.w;
    }
    __syncthreads();

    const int mr = mtile * 16 + m;
    const int nn = ntile * 16 + m;
#pragma unroll 4
    for (int ks = 0; ks < KC / 4; ++ks) {
      int kk = ks * 4 + 2 * hi;
      v2f ah = *(const v2f*)&Hs[mr][kk];
      v2f bre, bim;
      bre.x = Bs_re[kk + 0][nn];
      bre.y = Bs_re[kk + 1][nn];
      bim.x = Bs_im[kk + 0][nn];
      bim.y = Bs_im[kk + 1][nn];
      cre = __builtin_amdgcn_wmma_f32_16x16x4_f32(false, ah, false, bre, (short)0, cre, false, false);
      cim = __builtin_amdgcn_wmma_f32_16x16x4_f32(false, ah, false, bim, (short)0, cim, false, false);
    }
  }

  const int col = ntile * 16 + (lane & 15);
#pragma unroll
  for (int r = 0; r < 8; ++r) {
    int row = o0 + mtile * 16 + r + 8 * hi;
    size_t oo = f_base + (size_t)row * B_DIM + col;
    out_re[oo] = cre[r];
    out_im[oo] = cim[r];
  }
}

// ---------------------------------------------------------------------------
// irfft2 + bias.  One (b,o) image per workgroup, input layout (F, O, B).
// ---------------------------------------------------------------------------
__global__ __launch_bounds__(256) void k_irfft(const float* __restrict__ yr,
                                               const float* __restrict__ yi,
                                               const float* __restrict__ bias,
                                               float* __restrict__ out) {
  const int b = blockIdx.x >> 8;
  const int o = blockIdx.x & 255;
  __shared__ float twc[32], tws[32];
  __shared__ float ar[32][17], ai[32][17];
  __shared__ float zr[32][17], zi[32][17];
  const int tid = (int)threadIdx.x;
  if (tid < 32) {
    float s, co;
    __sincosf(TWOPI_32 * (float)tid, &s, &co);
    twc[tid] = co;
    tws[tid] = s;
  }
  for (int idx = tid; idx < 544; idx += 256) {
    size_t g = (size_t)idx * (C_DIM * B_DIM) + (size_t)o * B_DIM + b;
    ar[idx / 17][idx % 17] = yr[g];
    ai[idx / 17][idx % 17] = yi[g];
  }
  __syncthreads();
  // inverse FFT along u:  z[p][v] = (1/32) sum_u a[u][v] e^{+2pi i u p/32}
  for (int idx = tid; idx < 544; idx += 256) {
    int p = idx / 17, v = idx % 17;
    float sr = 0.0f, si = 0.0f;
#pragma unroll 4
    for (int u = 0; u < 32; ++u) {
      int t = (u * p) & 31;
      float cc = twc[t], ss = tws[t];
      sr += ar[u][v] * cc - ai[u][v] * ss;
      si += ar[u][v] * ss + ai[u][v] * cc;
    }
    zr[p][v] = sr * (1.0f / 32.0f);
    zi[p][v] = si * (1.0f / 32.0f);
  }
  __syncthreads();
  const float bv = bias[o];
  // inverse rFFT along v (hermitian):  y[p][q]
  for (int idx = tid; idx < 1024; idx += 256) {
    int p = idx >> 5, q = idx & 31;
    float acc = zr[p][0] + ((q & 1) ? -zr[p][16] : zr[p][16]);
#pragma unroll 4
    for (int v = 1; v < 16; ++v) {
      int t = (v * q) & 31;
      acc += 2.0f * (zr[p][v] * twc[t] - zi[p][v] * tws[t]);
    }
    out[(size_t)(b * 256 + o) * 1024 + idx] = acc * (1.0f / 32.0f) + bv;
  }
}

// ---------------------------------------------------------------------------
static inline int imin_i(int a, int b) { return a < b ? a : b; }

extern "C" void kernel_launch(void* const* d_in, const int* in_sizes, int n_in,
                              void* d_out, int out_size, void* d_ws, size_t ws_size,
                              hipStream_t stream) {
  (void)in_sizes; (void)n_in; (void)out_size;
  const float* x     = (const float*)d_in[0];   // (32,256,32,32)
  const float* w     = (const float*)d_in[1];   // (256,256,3,3)
  const float* alpha = (const float*)d_in[2];   // (1,)
  const float* Hm    = (const float*)d_in[3];   // (256,256,1,1)
  const float* bias  = (const float*)d_in[4];   // (256,)
  float* out = (float*)d_out;
  float* ws = (float*)d_ws;

  const size_t FCB = (size_t)NFREQ * C_DIM * B_DIM;  // 4,456,448 floats
  float* norm2 = ws;          // [0] = ||wfft||^2 accum, [1] = scale
  float* scale = ws + 1;
  float* xr = ws + 64;
  float* xi = xr + FCB;
  float* wxr = xi + FCB;
  float* wxi = wxr + FCB;
  float* chunkBase = wxi + FCB;

  const size_t persistentF = 64 + 4 * FCB;
  const size_t perF = 2 * (size_t)C_DIM * C_DIM + 4 * (size_t)C_DIM * B_DIM;  // floats/freq
  size_t availF = (ws_size / 4 > persistentF) ? (ws_size / 4 - persistentF) : 0;
  int chunkF = (int)(availF / perF);
  if (chunkF < 1) chunkF = 1;
  if (chunkF > NFREQ) chunkF = NFREQ;

  hipMemsetAsync(norm2, 0, 2 * sizeof(float), stream);
  k_xfft<<<32 * 256, 256, 0, stream>>>(x, xr, xi);
  k_norm<<<(NFREQ * 65536) / 256, 256, 0, stream>>>(w, norm2);
  k_scale<<<1, 1, 0, stream>>>(alpha, norm2, scale);

  float* Ar  = chunkBase;
  float* Ai  = Ar + (size_t)chunkF * 65536;
  float* t1r = Ai + (size_t)chunkF * 65536;
  float* t1i = t1r + (size_t)chunkF * 8192;
  float* t2r = t1i + (size_t)chunkF * 8192;
  float* t2i = t2r + (size_t)chunkF * 8192;

  for (int f0 = 0; f0 < NFREQ; f0 += chunkF) {
    int nf = imin_i(chunkF, NFREQ - f0);
    int nblk = (int)(((size_t)nf * 65536 + 255) / 256);
    k_buildA<<<nblk, 256, 0, stream>>>(w, scale, Ar, Ai, f0, nf);
    dim3 g((unsigned)nf, 4);
    // Horner: t1 = x - A x; t2 = x - A t1; t1 = x - A t2; wx = x - 2 A t1
    k_capply<<<g, 256, 0, stream>>>(Ar, Ai, xr, xi, xr, xi, t1r, t1i, f0, 1, 0, -1.0f);
    k_capply<<<g, 256, 0, stream>>>(Ar, Ai, xr, xi, t1r, t1i, t2r, t2i, f0, 0, 0, -1.0f);
    k_capply<<<g, 256, 0, stream>>>(Ar, Ai, xr, xi, t2r, t2i, t1r, t1i, f0, 0, 0, -1.0f);
    k_capply<<<g, 256, 0, stream>>>(Ar, Ai, xr, xi, t1r, t1i, wxr, wxi, f0, 0, 1, -2.0f);
  }

  // frequency-domain 1x1 conv (reuses xfft buffers as output)
  k_mixH<<<dim3(NFREQ, 4), 256, 0, stream>>>(Hm, wxr, wxi, xr, xi);
  // inverse transform + bias
  k_irfft<<<32 * 256, 256, 0, stream>>>(xr, xi, bias, out);
}